// FramePredictor_180388627259
// MI455X (gfx1250) — compile-verified
//
#include <hip/hip_runtime.h>

typedef float f32x4 __attribute__((ext_vector_type(4)));

namespace {
constexpr int kB = 4, kH = 1080, kW = 1920;
constexpr int kHP = kH + 2, kWP = kW + 2;               // padded splat grid 1082 x 1922
constexpr long long kHW    = (long long)kH * kW;        // 2,073,600
constexpr long long kNPix  = (long long)kB * kHW;       // 8,294,400
constexpr long long kNCell = (long long)kB * kHP * kWP; // 8,318,416 f32x4 cells (~127 MB)
constexpr float kMaxX = (float)(kW + 1);
constexpr float kMaxY = (float)(kH + 1);
} // namespace

#if __has_builtin(__builtin_amdgcn_global_store_async_from_lds_b128)
#define USE_ASYNC_ZERO 1
// Builtin signature (from hipcc diagnostic): param0 is
//   int __attribute__((vector_size(16))) __device__ *   (AS1 = HIP __device__)
// so pass exact vector-pointer types in the right address spaces.
typedef int v4i_ __attribute__((vector_size(16)));
typedef __attribute__((address_space(1))) v4i_* gp4i_t;
typedef __attribute__((address_space(3))) v4i_* lp4i_t;
#else
#define USE_ASYNC_ZERO 0
#endif

// ---------------------------------------------------------------------------
// Kernel 0: zero the (b, h+2, w+2, 4) accumulator and the global-max slot.
// On gfx1250 the zero-fill is issued through the async LDS->global DMA path
// (GLOBAL_STORE_ASYNC_FROM_LDS_B128, tracked on ASYNCcnt; S_ENDPGM's implicit
// S_WAIT_IDLE drains it). Zeroed lines land in L2 and stay resident for the
// atomic splat phase (127 MB accumulator < 192 MB L2).
// ---------------------------------------------------------------------------
__global__ void __launch_bounds__(256)
k_init(f32x4* __restrict__ acc, unsigned int* __restrict__ maxbits) {
#if USE_ASYNC_ZERO
  __shared__ v4i_ zsrc[32];
  if (threadIdx.x < 32) {
    v4i_ z = {0, 0, 0, 0};
    zsrc[threadIdx.x] = z;
  }
  __syncthreads();
#endif

  long long stride = (long long)gridDim.x * blockDim.x;
  long long i = (long long)blockIdx.x * blockDim.x + threadIdx.x;
  if (i == 0) *maxbits = 0u;
  for (; i < kNCell; i += stride) {
#if USE_ASYNC_ZERO
    __builtin_amdgcn_global_store_async_from_lds_b128(
        (gp4i_t)(acc + i), (lp4i_t)&zsrc[threadIdx.x & 31], 0, 0);
#else
    f32x4 z = {0.f, 0.f, 0.f, 0.f};
    acc[i] = z;
#endif
  }
}

// ---------------------------------------------------------------------------
// Kernel 1: global max of log1p(clip(depth, 0, 1000)) over all B*H*W values.
// wave32 shuffle tree -> LDS -> one atomicMax on uint bits (values >= 0, so
// float order == uint order). NT loads + global_prefetch_b8: pure streaming
// pass that must not pollute L2.
// ---------------------------------------------------------------------------
__global__ void __launch_bounds__(256)
k_maxlog(const float* __restrict__ depth, unsigned int* __restrict__ maxbits) {
  long long stride = (long long)gridDim.x * blockDim.x;
  long long i = (long long)blockIdx.x * blockDim.x + threadIdx.x;
  float m = 0.f;
  for (; i < kNPix; i += stride) {
    if (i + stride < kNPix) __builtin_prefetch(&depth[i + stride], 0, 0); // global_prefetch_b8
    float d = __builtin_nontemporal_load(&depth[i]);
    d = fminf(fmaxf(d, 0.f), 1000.f);
    m = fmaxf(m, log1pf(d));
  }
  #pragma unroll
  for (int off = 16; off > 0; off >>= 1) m = fmaxf(m, __shfl_xor(m, off, 32));
  __shared__ float smem[8]; // 256 threads / 32 lanes = 8 waves
  int lane = threadIdx.x & 31, wid = threadIdx.x >> 5;
  if (lane == 0) smem[wid] = m;
  __syncthreads();
  if (threadIdx.x < 8) {
    m = smem[threadIdx.x];
    #pragma unroll
    for (int off = 4; off > 0; off >>= 1) m = fmaxf(m, __shfl_xor(m, off, 32));
    if (threadIdx.x == 0) atomicMax(maxbits, __float_as_uint(m));
  }
}

// ---------------------------------------------------------------------------
// Kernel 2: bilinear splat. One thread per source pixel; 16 hardware
// global_atomic_add_f32 into the L2-resident accumulator. All streaming input
// reads are non-temporal so the accumulator stays hot in L2.
// 1/depth_w is folded into the exponent: exp(-log1p(d) * 50/maxlog).
// ---------------------------------------------------------------------------
__global__ void __launch_bounds__(256)
k_splat(const float* __restrict__ frame1, const float* __restrict__ depth,
        const float* __restrict__ flow, const unsigned int* __restrict__ maxbits,
        float* __restrict__ acc) {
  const int b = blockIdx.y;
  const long long rem = (long long)blockIdx.x * blockDim.x + threadIdx.x;
  if (rem >= kHW) return;
  const int r32 = (int)rem;
  const int y = r32 / kW;           // constant divisor -> mul/shift
  const int x = r32 - y * kW;

  const float scale = 50.f / __uint_as_float(*maxbits); // uniform scalar load

  const long long pix = (long long)b * kHW + rem;
  const float flx = __builtin_nontemporal_load(&flow[(long long)(2 * b) * kHW + rem]);
  const float fly = __builtin_nontemporal_load(&flow[(long long)(2 * b + 1) * kHW + rem]);
  const float d   = __builtin_nontemporal_load(&depth[pix]);
  const float invw = expf(-log1pf(fminf(fmaxf(d, 0.f), 1000.f)) * scale);

  const float* f1 = frame1 + (long long)b * 3 * kHW + rem;
  const float c0 = __builtin_nontemporal_load(&f1[0]);
  const float c1 = __builtin_nontemporal_load(&f1[kHW]);
  const float c2 = __builtin_nontemporal_load(&f1[2 * kHW]);

  const float tpox = flx + (float)x + 1.f;
  const float tpoy = fly + (float)y + 1.f;
  const float fxf = fminf(fmaxf(floorf(tpox), 0.f), kMaxX);
  const float fyf = fminf(fmaxf(floorf(tpoy), 0.f), kMaxY);
  const float cxf = fminf(fmaxf(ceilf(tpox),  0.f), kMaxX);
  const float cyf = fminf(fmaxf(ceilf(tpoy),  0.f), kMaxY);
  const float frx = fminf(fmaxf(tpox, 0.f), kMaxX) - fxf;
  const float fry = fminf(fmaxf(tpoy, 0.f), kMaxY) - fyf;

  const int fxi = (int)fxf, fyi = (int)fyf, cxi = (int)cxf, cyi = (int)cyf;
  const float wnw = (1.f - fry) * (1.f - frx) * invw;
  const float wsw = fry * (1.f - frx) * invw;
  const float wne = (1.f - fry) * frx * invw;
  const float wse = fry * frx * invw;

  float* accb = acc + (long long)b * kHP * kWP * 4;
  #define SPLAT1(IY, IX, WW)                                                  \
    {                                                                         \
      float* cell = accb + (((long long)(IY)) * kWP + (IX)) * 4;              \
      unsafeAtomicAdd(cell + 0, c0 * (WW));                                   \
      unsafeAtomicAdd(cell + 1, c1 * (WW));                                   \
      unsafeAtomicAdd(cell + 2, c2 * (WW));                                   \
      unsafeAtomicAdd(cell + 3, (WW));                                        \
    }
  SPLAT1(fyi, fxi, wnw);
  SPLAT1(cyi, fxi, wsw);
  SPLAT1(fyi, cxi, wne);
  SPLAT1(cyi, cxi, wse);
  #undef SPLAT1
}

// ---------------------------------------------------------------------------
// Kernel 3: normalize + clip + infill + mask, 4 pixels per thread.
// Accumulator cells are contiguous 128-bit loads (hit L2); outputs are
// write-once 128-bit NT stores; frame2 read only when a quad has an invalid
// pixel. out layout: [infilled (B,3,H,W)] ++ [mask (B,1,H,W)]
// ---------------------------------------------------------------------------
__global__ void __launch_bounds__(256)
k_out(const f32x4* __restrict__ acc, const float* __restrict__ frame2,
      float* __restrict__ out) {
  const int b = blockIdx.y;
  const long long q = (long long)blockIdx.x * blockDim.x + threadIdx.x;
  const long long rem = q * 4;
  if (rem >= kHW) return;
  const int r32 = (int)rem;
  const int y = r32 / kW;
  const int x = r32 - y * kW;   // x..x+3 stay in one row (kW % 4 == 0)

  const f32x4* row = acc + ((long long)b * kHP + (y + 1)) * kWP + (x + 1);
  f32x4 cells[4];
  #pragma unroll
  for (int j = 0; j < 4; ++j) cells[j] = row[j];

  const long long base = (long long)b * 3 * kHW + rem;

  bool anyInvalid = false;
  #pragma unroll
  for (int j = 0; j < 4; ++j) anyInvalid |= !(cells[j].w > 0.f);

  f32x4 fA = {0.f, 0.f, 0.f, 0.f}, fBv = fA, fCv = fA;
  if (anyInvalid) {
    fA  = __builtin_nontemporal_load((const f32x4*)&frame2[base]);
    fBv = __builtin_nontemporal_load((const f32x4*)&frame2[base + kHW]);
    fCv = __builtin_nontemporal_load((const f32x4*)&frame2[base + 2 * kHW]);
  }

  f32x4 oA, oB, oC, mv;
  #pragma unroll
  for (int j = 0; j < 4; ++j) {
    const f32x4 c = cells[j];
    if (c.w > 0.f) {
      const float inv = 1.f / c.w;
      oA[j] = fminf(fmaxf(c.x * inv, -1.f), 1.f);
      oB[j] = fminf(fmaxf(c.y * inv, -1.f), 1.f);
      oC[j] = fminf(fmaxf(c.z * inv, -1.f), 1.f);
      mv[j] = 1.f;
    } else {
      oA[j] = fA[j];
      oB[j] = fBv[j];
      oC[j] = fCv[j];
      mv[j] = 0.f;
    }
  }

  __builtin_nontemporal_store(oA, (f32x4*)&out[base]);
  __builtin_nontemporal_store(oB, (f32x4*)&out[base + kHW]);
  __builtin_nontemporal_store(oC, (f32x4*)&out[base + 2 * kHW]);
  __builtin_nontemporal_store(mv, (f32x4*)&out[(long long)kB * 3 * kHW + (long long)b * kHW + rem]);
}

// ---------------------------------------------------------------------------
extern "C" void kernel_launch(void* const* d_in, const int* in_sizes, int n_in,
                              void* d_out, int out_size, void* d_ws, size_t ws_size,
                              hipStream_t stream) {
  const float* frame1 = (const float*)d_in[0]; // (B,3,H,W)
  const float* depth1 = (const float*)d_in[1]; // (B,1,H,W)
  const float* flow12 = (const float*)d_in[2]; // (B,2,H,W)
  const float* frame2 = (const float*)d_in[3]; // (1,B,3,H,W) -> flat (B,3,H,W)
  float* out = (float*)d_out;

  // workspace: [kNCell f32x4 accumulator][1 uint global-max slot]
  float* acc = (float*)d_ws;
  unsigned int* maxbits = (unsigned int*)((char*)d_ws + (size_t)kNCell * sizeof(f32x4));

  const dim3 blk(256);
  const unsigned blocksCell = (unsigned)((kNCell + 255) / 256);
  const unsigned blocksHW   = (unsigned)((kHW + 255) / 256);       // 8100
  const unsigned blocksQuad = (unsigned)((kHW / 4 + 255) / 256);   // 2025

  k_init  <<<dim3(blocksCell), blk, 0, stream>>>((f32x4*)acc, maxbits);
  k_maxlog<<<dim3(1024),       blk, 0, stream>>>(depth1, maxbits);
  k_splat <<<dim3(blocksHW, kB), blk, 0, stream>>>(frame1, depth1, flow12, maxbits, acc);
  k_out   <<<dim3(blocksQuad, kB), blk, 0, stream>>>((const f32x4*)acc, frame2, out);
}